// DomainEncoder_11768210391115
// MI455X (gfx1250) — compile-verified
//
#include <hip/hip_runtime.h>

typedef __attribute__((ext_vector_type(2))) float v2f;
typedef __attribute__((ext_vector_type(8))) float v8f;
typedef __attribute__((address_space(3))) float lds_f;

#define N_TOK 32768
#define DIN 256
#define DHID 1024
#define DOUT 256
#define ND 8
#define ROWS 32
#define MAX_TILES (N_TOK / ROWS)
#define SX_STRIDE 260   // == 4 (mod 64 banks): conflict-free column reads
#define SH_STRIDE 1028  // == 4 (mod 64 banks)
#define LN_EPS 1e-5f

// ---------------- dispatch (bin tokens by domain) ----------------

__global__ void k_zero(int* w) {
  if (threadIdx.x < 16) w[threadIdx.x] = 0;  // counts[8] + cursors[8]
}

__global__ void k_count(const int* __restrict__ dom, int* __restrict__ counts, int n) {
  int i = blockIdx.x * blockDim.x + threadIdx.x;
  if (i < n) atomicAdd(&counts[dom[i]], 1);
}

__global__ void k_scan(const int* __restrict__ counts, int* __restrict__ offsets) {
  if (threadIdx.x == 0) {
    int a = 0;
    for (int d = 0; d < ND; ++d) { offsets[d] = a; a += counts[d]; }
  }
}

__global__ void k_scatter(const int* __restrict__ dom, const int* __restrict__ offsets,
                          int* __restrict__ cursors, int* __restrict__ perm, int n) {
  int i = blockIdx.x * blockDim.x + threadIdx.x;
  if (i < n) {
    int d = dom[i];
    int p = atomicAdd(&cursors[d], 1);
    perm[offsets[d] + p] = i;
  }
}

// Repack W[K][NC] (per domain) into P[kp][n][2] = {W[2kp][n], W[2kp+1][n]}
// so a WMMA B-fragment is one contiguous b64 per lane.
__global__ void k_pack(const float* __restrict__ W, float* __restrict__ P,
                       int K, int NC) {
  long i = (long)blockIdx.x * blockDim.x + threadIdx.x;
  long total = (long)ND * (K >> 1) * NC;
  if (i >= total) return;
  int n   = (int)(i % NC);
  long t  = i / NC;
  int kp  = (int)(t % (K >> 1));
  int dd  = (int)(t / (K >> 1));
  const float* src = W + ((size_t)dd * K + (size_t)(kp << 1)) * NC + n;
  v2f v;
  v.x = src[0];
  v.y = src[NC];
  *(v2f*)(P + (size_t)i * 2) = v;
}

// ---------------- fused expert MLP over one 32-row tile ----------------

template <bool PACKED>
__launch_bounds__(256, 1)
__global__ void k_mlp(const float* __restrict__ x,
                      const float* __restrict__ W1, const float* __restrict__ b1,
                      const float* __restrict__ gamma, const float* __restrict__ beta,
                      const float* __restrict__ W2, const float* __restrict__ b2,
                      const float* __restrict__ W1p, const float* __restrict__ W2p,
                      float* __restrict__ out,
                      const int* __restrict__ counts, const int* __restrict__ offsets,
                      const int* __restrict__ perm) {
  __shared__ float sX[ROWS * SX_STRIDE];
  __shared__ float sH[ROWS * SH_STRIDE];
  __shared__ int srow[ROWS];

  const int bid  = blockIdx.x;
  const int d    = bid & (ND - 1);
  const int tile = bid >> 3;
  const int cnt  = counts[d];
  const int row0 = tile * ROWS;
  if (row0 >= cnt) return;                 // uniform early-exit for surplus tiles
  const int base   = offsets[d];
  const int nvalid = min(ROWS, cnt - row0);

  const int tid = threadIdx.x;
  if (tid < ROWS) {
    int rr = (tid < nvalid) ? (row0 + tid) : row0;  // clamp padding rows to a valid token
    srow[tid] = perm[base + rr];
  }
  __syncthreads();

  // gather X tile (32 x 256 fp32) straight into LDS with async b128 copies
  {
    for (int f = tid; f < ROWS * (DIN / 4); f += 256) {
      int r  = f >> 6;      // 64 float4 per row
      int c4 = f & 63;
      const float* gsrc = x + (size_t)srow[r] * DIN + (c4 << 2);
      lds_f* ldst = (lds_f*)&sX[r * SX_STRIDE + (c4 << 2)];
      asm volatile("global_load_async_to_lds_b128 %0, %1, off"
                   :: "v"(ldst), "v"(gsrc) : "memory");
    }
    asm volatile("s_wait_asynccnt 0x0" ::: "memory");
  }
  __syncthreads();

  const int wave  = tid >> 5;
  const int lane  = tid & 31;
  const int nlo   = lane & 15;   // N index (and A-matrix row M) within a 16-tile
  const int khalf = lane >> 4;   // lanes 16-31 carry K+2/K+3 per ISA 16x4 layout

  // ---- stage 1: H = X @ W1[d] + b1[d]; this wave owns cols [wave*128, wave*128+128)
  {
    const float* W1d  = W1  + (size_t)d * DIN * DHID;
    const float* W1pd = W1p + (size_t)d * (DIN / 2) * DHID * 2;
    const float* b1d  = b1  + (size_t)d * DHID;
    const int cb = wave * 128;
    v8f acc[2][8];                               // [row-tile][col-tile]
#pragma unroll
    for (int nt = 0; nt < 8; ++nt) {
      float bb = b1d[cb + nt * 16 + nlo];
      v8f t = {bb, bb, bb, bb, bb, bb, bb, bb};
      acc[0][nt] = t;
      acc[1][nt] = t;
    }
    for (int k0 = 0; k0 < DIN; k0 += 4) {
      const int ka = k0 + 2 * khalf;
      v2f a0 = *(const v2f*)(&sX[nlo * SX_STRIDE + ka]);         // rows 0..15
      v2f a1 = *(const v2f*)(&sX[(nlo + 16) * SX_STRIDE + ka]);  // rows 16..31
#pragma unroll
      for (int nt = 0; nt < 8; ++nt) {
        const int nc = cb + nt * 16 + nlo;
        v2f b;
        if (PACKED) {
          b = *(const v2f*)(W1pd + (((size_t)(k0 >> 1) + khalf) * DHID + nc) * 2);
        } else {
          b.x = W1d[(size_t)ka * DHID + nc];
          b.y = W1d[(size_t)(ka + 1) * DHID + nc];
        }
        acc[0][nt] = __builtin_amdgcn_wmma_f32_16x16x4_f32(
            false, a0, false, b, (short)0, acc[0][nt], false, false);
        acc[1][nt] = __builtin_amdgcn_wmma_f32_16x16x4_f32(
            false, a1, false, b, (short)0, acc[1][nt], false, false);
      }
    }
#pragma unroll
    for (int rt = 0; rt < 2; ++rt)
#pragma unroll
      for (int nt = 0; nt < 8; ++nt)
#pragma unroll
        for (int v = 0; v < 8; ++v)
          sH[(rt * 16 + v + 8 * khalf) * SH_STRIDE + cb + nt * 16 + nlo] = acc[rt][nt][v];
  }
  __syncthreads();

  // ---- LayerNorm + ReLU (two-pass, matches reference); 4 rows per wave
  {
    const float* gd = gamma + (size_t)d * DHID;
    const float* bd = beta  + (size_t)d * DHID;
#pragma unroll
    for (int rr = 0; rr < 4; ++rr) {
      const int r = wave * 4 + rr;
      float s = 0.f;
      for (int c = lane; c < DHID; c += 32) s += sH[r * SH_STRIDE + c];
#pragma unroll
      for (int off = 16; off > 0; off >>= 1) s += __shfl_xor(s, off, 32);
      const float mu = s * (1.f / DHID);
      float vv = 0.f;
      for (int c = lane; c < DHID; c += 32) {
        float h = sH[r * SH_STRIDE + c] - mu;
        vv += h * h;
      }
#pragma unroll
      for (int off = 16; off > 0; off >>= 1) vv += __shfl_xor(vv, off, 32);
      const float inv = rsqrtf(vv * (1.f / DHID) + LN_EPS);
      for (int c = lane; c < DHID; c += 32) {
        float h = (sH[r * SH_STRIDE + c] - mu) * inv * gd[c] + bd[c];
        sH[r * SH_STRIDE + c] = fmaxf(h, 0.f);
      }
    }
  }
  __syncthreads();

  // ---- stage 2: OUT = H @ W2[d] + b2[d]; this wave owns cols [wave*32, wave*32+32)
  {
    const float* W2d  = W2  + (size_t)d * DHID * DOUT;
    const float* W2pd = W2p + (size_t)d * (DHID / 2) * DOUT * 2;
    const float* b2d  = b2  + (size_t)d * DOUT;
    const int cb = wave * 32;
    v8f acc[2][2];                               // [row-tile][col-tile]
#pragma unroll
    for (int nt = 0; nt < 2; ++nt) {
      float bb = b2d[cb + nt * 16 + nlo];
      v8f t = {bb, bb, bb, bb, bb, bb, bb, bb};
      acc[0][nt] = t;
      acc[1][nt] = t;
    }
    for (int k0 = 0; k0 < DHID; k0 += 4) {
      const int ka = k0 + 2 * khalf;
      v2f a0 = *(const v2f*)(&sH[nlo * SH_STRIDE + ka]);
      v2f a1 = *(const v2f*)(&sH[(nlo + 16) * SH_STRIDE + ka]);
#pragma unroll
      for (int nt = 0; nt < 2; ++nt) {
        const int nc = cb + nt * 16 + nlo;
        v2f b;
        if (PACKED) {
          b = *(const v2f*)(W2pd + (((size_t)(k0 >> 1) + khalf) * DOUT + nc) * 2);
        } else {
          b.x = W2d[(size_t)ka * DOUT + nc];
          b.y = W2d[(size_t)(ka + 1) * DOUT + nc];
        }
        acc[0][nt] = __builtin_amdgcn_wmma_f32_16x16x4_f32(
            false, a0, false, b, (short)0, acc[0][nt], false, false);
        acc[1][nt] = __builtin_amdgcn_wmma_f32_16x16x4_f32(
            false, a1, false, b, (short)0, acc[1][nt], false, false);
      }
    }
#pragma unroll
    for (int rt = 0; rt < 2; ++rt)
#pragma unroll
      for (int nt = 0; nt < 2; ++nt)
#pragma unroll
        for (int v = 0; v < 8; ++v) {
          const int mrow = rt * 16 + v + 8 * khalf;
          if (mrow < nvalid)
            out[(size_t)srow[mrow] * DOUT + cb + nt * 16 + nlo] = acc[rt][nt][v];
        }
  }
}

// ---------------- launcher ----------------

extern "C" void kernel_launch(void* const* d_in, const int* in_sizes, int n_in,
                              void* d_out, int out_size, void* d_ws, size_t ws_size,
                              hipStream_t stream) {
  const float* x     = (const float*)d_in[0];
  const int*   dom   = (const int*)d_in[1];
  const float* W1    = (const float*)d_in[2];
  const float* b1    = (const float*)d_in[3];
  const float* gamma = (const float*)d_in[4];
  const float* beta  = (const float*)d_in[5];
  const float* W2    = (const float*)d_in[6];
  const float* b2    = (const float*)d_in[7];
  float* out = (float*)d_out;

  int* wsi     = (int*)d_ws;
  int* counts  = wsi;        // 8
  int* cursors = wsi + 8;    // 8
  int* offsets = wsi + 16;   // 8
  int* perm    = wsi + 32;   // N_TOK

  const size_t packElems1 = (size_t)ND * (DIN / 2) * DHID * 2;   // 2M floats
  const size_t packElems2 = (size_t)ND * (DHID / 2) * DOUT * 2;  // 2M floats
  float* p1 = (float*)(wsi + 32 + N_TOK);
  float* p2 = p1 + packElems1;
  const size_t needed = (size_t)(32 + N_TOK) * 4 + (packElems1 + packElems2) * 4;
  const bool packed = (ws_size >= needed);

  k_zero<<<1, 32, 0, stream>>>(wsi);
  k_count<<<N_TOK / 256, 256, 0, stream>>>(dom, counts, N_TOK);
  k_scan<<<1, 32, 0, stream>>>(counts, offsets);
  k_scatter<<<N_TOK / 256, 256, 0, stream>>>(dom, offsets, cursors, perm, N_TOK);

  if (packed) {
    const long tot1 = (long)ND * (DIN / 2) * DHID;   // 1,048,576
    const long tot2 = (long)ND * (DHID / 2) * DOUT;  // 1,048,576
    k_pack<<<(int)((tot1 + 255) / 256), 256, 0, stream>>>(W1, p1, DIN, DHID);
    k_pack<<<(int)((tot2 + 255) / 256), 256, 0, stream>>>(W2, p2, DHID, DOUT);
    k_mlp<true><<<ND * MAX_TILES, 256, 0, stream>>>(
        x, W1, b1, gamma, beta, W2, b2, p1, p2, out, counts, offsets, perm);
  } else {
    k_mlp<false><<<ND * MAX_TILES, 256, 0, stream>>>(
        x, W1, b1, gamma, beta, W2, b2, p1, p2, out, counts, offsets, perm);
  }
}